// lstm_model_8873402433789
// MI455X (gfx1250) — compile-verified
//
#include <hip/hip_runtime.h>

typedef __attribute__((ext_vector_type(2))) float v2f;
typedef __attribute__((ext_vector_type(8))) float v8f;

#define B_ 8192
#define T_ 200
#define F_ 16
#define H_ 32
#define D_ 60
#define G_ 128   // 4*H

#define THREADS 64   // 2 waves cooperating on one 16-row batch tile

__device__ __forceinline__ float sigm(float v) {
    return 1.0f / (1.0f + __expf(-v));
}

__global__ __launch_bounds__(THREADS) void lstm_wmma_kernel(
    const float* __restrict__ x,  const float* __restrict__ W,
    const float* __restrict__ U,  const float* __restrict__ b,
    const float* __restrict__ W1, const float* __restrict__ b1,
    const float* __restrict__ W2, const float* __restrict__ b2,
    float* __restrict__ out)
{
    // W, U pre-swizzled into per-(n,c) 64-float blocks so a B-fragment is one
    // contiguous b64 per lane:  block[lane*2 + j] = Wt[K][N],
    // N = 16n + (lane&15), K = 4c + 2*(lane>>4) + j.
    __shared__ float Wlds[2048];              // 16x128
    __shared__ float Ulds[4096];              // 32x128
    __shared__ float hS[2][16][36];           // double-buffered h scratch (padded rows)

    const int tid = threadIdx.x;
    for (int i = tid; i < 2048; i += THREADS) {
        int n = i >> 8, rem = i & 255;
        int c = rem >> 6, q = rem & 63;
        int ln = q >> 1, j = q & 1;
        int K = 4 * c + 2 * (ln >> 4) + j;
        int N = 16 * n + (ln & 15);
        Wlds[i] = W[K * G_ + N];
    }
    for (int i = tid; i < 4096; i += THREADS) {
        int n = i >> 9, rem = i & 511;
        int c = rem >> 6, q = rem & 63;
        int ln = q >> 1, j = q & 1;
        int K = 4 * c + 2 * (ln >> 4) + j;
        int N = 16 * n + (ln & 15);
        Ulds[i] = U[K * G_ + N];
    }
    __syncthreads();

    const int p     = tid >> 5;     // wave id = gate column group j
    const int lane  = tid & 31;
    const int half  = lane >> 4;
    const int lan15 = lane & 15;
    const int b0    = blockIdx.x * 16;

    // wave p owns n-tiles n_m = 2m + p  (m=0:i_p, 1:f_p, 2:cc_p, 3:o_p)
    float bias[4];
#pragma unroll
    for (int m = 0; m < 4; ++m) bias[m] = b[16 * (2 * m + p) + lan15];

    v8f cst = {};          // c state for columns 16p..16p+15
    v2f hA[8];
#pragma unroll
    for (int c = 0; c < 8; ++c) hA[c] = (v2f){0.0f, 0.0f};

    const float* xrow = x + (size_t)(b0 + lan15) * T_ * F_;

    for (int t = 0; t < T_; ++t) {
        // A-fragments of x_t: lane<16 holds K=4c,4c+1 of row lan15; lane>=16 K=4c+2,4c+3
        v2f xA[4];
        const float* xp = xrow + (size_t)t * F_ + 2 * half;
#pragma unroll
        for (int c = 0; c < 4; ++c)
            xA[c] = *(const v2f*)(xp + 4 * c);

        v8f acc[4];
#pragma unroll
        for (int m = 0; m < 4; ++m) {
            int n = 2 * m + p;
            float bb = bias[m];
            v8f a = (v8f){bb, bb, bb, bb, bb, bb, bb, bb};
            // g += x_t @ W   (K = 16 -> 4 chunks)
#pragma unroll
            for (int c = 0; c < 4; ++c) {
                v2f wb = *(const v2f*)&Wlds[(n * 4 + c) * 64 + lane * 2];
                a = __builtin_amdgcn_wmma_f32_16x16x4_f32(
                        false, xA[c], false, wb, (short)0, a, false, false);
            }
            // g += h @ U     (K = 32 -> 8 chunks)
#pragma unroll
            for (int c = 0; c < 8; ++c) {
                v2f ub = *(const v2f*)&Ulds[(n * 8 + c) * 64 + lane * 2];
                a = __builtin_amdgcn_wmma_f32_16x16x4_f32(
                        false, hA[c], false, ub, (short)0, a, false, false);
            }
            acc[m] = a;
        }

        // Keras order i,f,cc,o : c = sig(f)*c + sig(i)*relu(cc); h = sig(o)*relu(c)
        v8f hh;
#pragma unroll
        for (int e = 0; e < 8; ++e) {
            float iv = sigm(acc[0][e]);
            float fv = sigm(acc[1][e]);
            float cc = fmaxf(acc[2][e], 0.0f);
            float ov = sigm(acc[3][e]);
            float cn = fv * cst[e] + iv * cc;
            cst[e] = cn;
            hh[e] = ov * fmaxf(cn, 0.0f);
        }

        // C-layout h tile -> LDS (row = e + 8*half, col = 16p + lan15)
        float (*hb)[36] = hS[t & 1];
#pragma unroll
        for (int e = 0; e < 8; ++e)
            hb[e + 8 * half][16 * p + lan15] = hh[e];

        __syncthreads();   // single barrier: ping-pong buffer protects reads vs next writes

        // LDS -> A-layout h fragments (lane<16: K=4c,4c+1 ; lane>=16: K=4c+2,4c+3)
#pragma unroll
        for (int c = 0; c < 8; ++c)
            hA[c] = *(const v2f*)&hb[lan15][4 * c + 2 * half];
    }

    // Heads: long = h@W1+b1, lat = h@W2+b2 ; final h is in hS[(T_-1)&1].
    float (*hs)[36] = hS[(T_ - 1) & 1];
    for (int idx = tid; idx < 16 * D_; idx += THREADS) {   // 960 % 64 == 0 -> uniform
        int r = idx / D_;
        int d = idx - r * D_;
        float s1 = b1[d];
        float s2 = b2[d];
#pragma unroll
        for (int k = 0; k < H_; ++k) {
            float hv = hs[r][k];
            s1 += hv * W1[k * D_ + d];
            s2 += hv * W2[k * D_ + d];
        }
        size_t row = (size_t)(b0 + r);
        out[row * D_ + d]                   = s1;
        out[(size_t)B_ * D_ + row * D_ + d] = s2;
    }
}

extern "C" void kernel_launch(void* const* d_in, const int* in_sizes, int n_in,
                              void* d_out, int out_size, void* d_ws, size_t ws_size,
                              hipStream_t stream) {
    (void)in_sizes; (void)n_in; (void)d_ws; (void)ws_size; (void)out_size;
    const float* x  = (const float*)d_in[0];
    const float* W  = (const float*)d_in[1];
    const float* U  = (const float*)d_in[2];
    const float* b  = (const float*)d_in[3];
    const float* W1 = (const float*)d_in[4];
    const float* b1 = (const float*)d_in[5];
    const float* W2 = (const float*)d_in[6];
    const float* b2 = (const float*)d_in[7];
    float* out = (float*)d_out;

    dim3 grid(B_ / 16);   // 512 blocks x 2 waves = 1024 waves
    dim3 block(THREADS);
    lstm_wmma_kernel<<<grid, block, 0, stream>>>(x, W, U, b, W1, b1, W2, b2, out);
}